// Lowpass_45097156608496
// MI455X (gfx1250) — compile-verified
//
#include <hip/hip_runtime.h>
#include <stdint.h>

#define B_ 16
#define T_ 2048
#define U_ 1024
#define NSEG 8
#define SEG_LEN 256           // T_/NSEG
#define CHUNKS 16             // SEG_LEN/16
#define UBLK 32               // units per workgroup == waves per workgroup
#define BLKDW 256             // dwords per (unit, buffer) LDS block (16b x 16t)
#define BUFDW (UBLK*BLKDW)    // 8192 dwords per buffer

typedef float v2f __attribute__((ext_vector_type(2)));
typedef float v8f __attribute__((ext_vector_type(8)));

// s^e for e in [0,15], sharing precomputed squarings (4 cndmask+mul pairs).
__device__ __forceinline__ float ipow16(float s, float s2, float s4, float s8,
                                        int e) {
  float r = (e & 1) ? s : 1.0f;
  r *= (e & 2) ? s2 : 1.0f;
  r *= (e & 4) ? s4 : 1.0f;
  r *= (e & 8) ? s8 : 1.0f;
  return r;
}

// ---------------------------------------------------------------------------
// Phase 1: per-segment local scan via WMMA.
// Workgroup = 32 waves; wave w owns unit u = u0 + w.
// Per 16-step chunk: D[b,t] = sum_k X[b,k] * Acoef[t,k] + s^{t+1} * carry[b]
//   A-frag (16x4 f32): lane holds row m=b=lane%16, vgpr j holds K = j + 2*(lane/16)
//   B-frag (4x16 f32): lane holds col n=t=lane%16, vgpr j holds K-row = j + 2*(lane/16)
//   C/D    (16x16)   : lane holds n=lane%16, vgpr i holds m = i + 8*(lane/16)
// ---------------------------------------------------------------------------
extern "C" __global__ __launch_bounds__(1024, 1)
void lowpass_scan_wmma(const float* __restrict__ in,
                       const float* __restrict__ level0,
                       const float* __restrict__ smooth,
                       float* __restrict__ out,
                       float* __restrict__ Eseg) {
  __shared__ float sbuf[2 * BUFDW];   // 64 KB, double buffered

  const int tid  = threadIdx.x;
  const int w    = tid >> 5;          // wave id = local unit index
  const int lane = tid & 31;
  const int th   = lane >> 4;         // lane half
  const int bl   = lane & 15;
  const int u0   = blockIdx.x * UBLK;
  const int j    = blockIdx.y;        // segment index
  const int u    = u0 + w;
  const int seg0 = j * SEG_LEN;

  // per-unit smoothing coefficient (uniform within wave)
  const float s   = 1.0f / (1.0f + expf(-smooth[u]));
  const float oms = 1.0f - s;
  const float s2 = s * s, s4 = s2 * s2, s8 = s4 * s4;

  // B-fragments: coef[t=bl, k=4*kk+jj+2*th] = (k<=t) ? (1-s)*s^(t-k) : 0
  v2f cb[4];
#pragma unroll
  for (int kk = 0; kk < 4; ++kk) {
#pragma unroll
    for (int jj = 0; jj < 2; ++jj) {
      const int kr = 4 * kk + jj + 2 * th;
      cb[kk][jj] = (kr <= bl)
          ? oms * ipow16(s, s2, s4, s8, (bl - kr) & 15) : 0.0f;
    }
  }
  const float pw = s * ipow16(s, s2, s4, s8, bl);  // s^(t+1), t = bl

  // carry per half-wave: ch[i] = carry for batch row b = i + 8*th
  float ch[8];
  const float cinit = (j == 0) ? level0[u] : 0.0f;  // later segments fixed up in phase 2
#pragma unroll
  for (int i = 0; i < 8; ++i) ch[i] = cinit;

  // ---- loop-invariant addressing ------------------------------------------
  // Wave w's 8 async rows: fixed batch row bb = w>>1, tt = (w&1)*8 + q.
  const int bb_w = w >> 1;
  const int tt0  = (w & 1) * 8;
  // per-lane 64-bit global base; +q*4096 per row, +65536 per chunk
  uint64_t gaddr = (uint64_t)(uintptr_t)
      (in + (((size_t)bb_w * T_ + seg0 + tt0) * U_ + u0 + lane));
  // per-row LDS destination byte offsets, both buffers precomputed
  unsigned loffq0[8], loffq1[8];
#pragma unroll
  for (int q = 0; q < 8; ++q) {
    loffq0[q] = 4u * (unsigned)(lane * BLKDW + (((tt0 + q) * 16 + bb_w) ^ lane));
    loffq1[q] = loffq0[q] + (unsigned)(BUFDW * 4);
  }
  // A-fragment LDS read indices (buffer 0); +BUFDW selects buffer 1
  int rbase[8];
#pragma unroll
  for (int kk = 0; kk < 4; ++kk)
#pragma unroll
    for (int jj = 0; jj < 2; ++jj)
      rbase[kk * 2 + jj] =
          w * BLKDW + ((((4 * kk + jj + 2 * th) * 16) + bl) ^ w);
  // stores: uniform (SGPR) segment base + per-batch-row 32-bit offsets;
  // the per-chunk term c*16*U_*4 (<= 960KB) folds into immediates.
  float* const outseg = out + (size_t)seg0 * U_;
  int voff[8];
#pragma unroll
  for (int i = 0; i < 8; ++i)
    voff[i] = ((i + 8 * th) * T_ + bl) * U_ + u;

  // volatile LDS view: guarantees real ds loads (sbuf only written via asm).
  volatile const float* sb = sbuf;

  // ---- async tile issue ----------------------------------------------------
  auto issue = [&](const unsigned* loffs) {
#pragma unroll
    for (int q = 0; q < 8; ++q) {
      const uint64_t ga = gaddr + (uint64_t)(q * 4096);
      asm volatile("global_load_async_to_lds_b32 %0, %1, off"
                   :: "v"(loffs[q]), "v"(ga)
                   : "memory");
    }
    gaddr += 16u * U_ * 4u;   // advance one chunk
  };

  issue(loffq0);
  asm volatile("s_wait_asynccnt 0" ::: "memory");
  __syncthreads();

#pragma unroll
  for (int c = 0; c < CHUNKS; ++c) {
    const int sel = c & 1;
    if (c + 1 < CHUNKS) issue(sel ? loffq0 : loffq1);

    // accumulator starts as carry contribution: C[b,t] = s^{t+1} * carry[b]
    v8f acc;
#pragma unroll
    for (int i = 0; i < 8; ++i) acc[i] = pw * ch[i];

#pragma unroll
    for (int kk = 0; kk < 4; ++kk) {
      v2f a;
#pragma unroll
      for (int jj = 0; jj < 2; ++jj)
        a[jj] = sb[sel * BUFDW + rbase[kk * 2 + jj]];
      acc = __builtin_amdgcn_wmma_f32_16x16x4_f32(
          false, a, false, cb[kk], (short)0, acc, false, false);
    }

    // store: lane column n=bl -> time, vgpr i -> batch row b=i+8*th
#pragma unroll
    for (int i = 0; i < 8; ++i)
      outseg[(size_t)voff[i] + (size_t)(c * 16 * U_)] = acc[i];

    // carry update: column t=15 lives in lane 15 (b=0..7) / lane 31 (b=8..15)
#pragma unroll
    for (int i = 0; i < 8; ++i)
      ch[i] = __shfl(acc[i], 15 + 16 * th, 32);

    asm volatile("s_wait_asynccnt 0" ::: "memory");
    __syncthreads();
  }

  // record local end-of-segment levels for the cross-segment fixup
  if (bl == 0) {
#pragma unroll
    for (int i = 0; i < 8; ++i)
      Eseg[((size_t)j * B_ + (th * 8 + i)) * U_ + u] = ch[i];
  }
}

// ---------------------------------------------------------------------------
// Phase 2: fold the tiny cross-segment carry chain and apply
//   out[b, j*L + t, u] += s^{t+1} * Lend[j-1, b, u]
// Streamed with float4 loads/stores along u.
// ---------------------------------------------------------------------------
extern "C" __global__ __launch_bounds__(256, 1)
void lowpass_fixup(float* __restrict__ out,
                   const float* __restrict__ smooth,
                   const float* __restrict__ Eseg) {
  const int j  = blockIdx.x + 1;   // segments 1..NSEG-1
  const int b  = blockIdx.y;
  const int tq = blockIdx.z;       // 64-step slice of the segment
  const int u  = threadIdx.x * 4;

  float s[4], sL[4], c[4], pwv[4];
#pragma unroll
  for (int k = 0; k < 4; ++k) {
    s[k] = 1.0f / (1.0f + expf(-smooth[u + k]));
    // s^256 by 8 squarings
    float p = s[k];
#pragma unroll
    for (int r = 0; r < 8; ++r) p *= p;
    sL[k] = p;
  }
  // Lend[j-1] = E[j-1] + sL*(E[j-2] + sL*(... E[0]))   (E[0] includes level_var)
#pragma unroll
  for (int k = 0; k < 4; ++k) c[k] = Eseg[(size_t)b * U_ + u + k];
  for (int jj = 1; jj < j; ++jj) {
#pragma unroll
    for (int k = 0; k < 4; ++k)
      c[k] = Eseg[((size_t)jj * B_ + b) * U_ + u + k] + sL[k] * c[k];
  }

  // pw = s^(tq*64 + 1): pick s^(64*tq) from squarings, then * s
#pragma unroll
  for (int k = 0; k < 4; ++k) {
    float p = s[k];
#pragma unroll
    for (int r = 0; r < 6; ++r) p *= p;      // s^64
    const float p128 = p * p;
    float sel = 1.0f;
    if (tq == 1) sel = p;
    else if (tq == 2) sel = p128;
    else if (tq == 3) sel = p128 * p;
    pwv[k] = sel * s[k];
  }

  const int t0 = tq * 64;
  float4* o4 = (float4*)(out + ((size_t)b * T_ + j * SEG_LEN + t0) * U_ + u);
  for (int t = 0; t < 64; ++t) {
    float4 v = o4[t * (U_ / 4)];
    v.x += pwv[0] * c[0];
    v.y += pwv[1] * c[1];
    v.z += pwv[2] * c[2];
    v.w += pwv[3] * c[3];
    o4[t * (U_ / 4)] = v;
#pragma unroll
    for (int k = 0; k < 4; ++k) pwv[k] *= s[k];
  }
}

extern "C" void kernel_launch(void* const* d_in, const int* in_sizes, int n_in,
                              void* d_out, int out_size, void* d_ws, size_t ws_size,
                              hipStream_t stream) {
  const float* in = (const float*)d_in[0];   // [B,T,U] fp32
  const float* lv = (const float*)d_in[1];   // [1,U]
  const float* sm = (const float*)d_in[2];   // [1,U]
  float* out = (float*)d_out;                // [B,T,U]
  float* E   = (float*)d_ws;                 // NSEG*B*U floats = 512 KB scratch

  dim3 g1(U_ / UBLK, NSEG);                  // 32 x 8 = 256 workgroups
  lowpass_scan_wmma<<<g1, dim3(1024, 1, 1), 0, stream>>>(in, lv, sm, out, E);

  dim3 g2(NSEG - 1, B_, 4);                  // 7 x 16 x 4 = 448 workgroups
  lowpass_fixup<<<g2, dim3(256, 1, 1), 0, stream>>>(out, sm, E);
}